// LogAbsSumDet_69526930587790
// MI455X (gfx1250) — compile-verified
//
#include <hip/hip_runtime.h>
#include <cstdint>
#include <cstddef>

typedef float v2f __attribute__((ext_vector_type(2)));
typedef float v8f __attribute__((ext_vector_type(8)));

#define NDET 16
#define NDIM 16

// One wave (32 lanes) per sample:
//   lanes 0..15  : LU of a[s, d]   (d = lane)
//   lanes 16..31 : LU of b[s, d]   (d = lane-16)
// Each lane holds its whole 16x16 matrix in registers and does a fully
// unrolled no-pivot LU (batched-tiny-LU, one-matrix-per-thread style).
// The weighted signed-LSE over the 16 determinants is done with four
// chained V_WMMA_F32_16X16X4_F32 rank-4 accumulations.
__global__ __launch_bounds__(128, 1)
void logabssumdet_kernel(const float* __restrict__ A,
                         const float* __restrict__ B,
                         const float* __restrict__ W,
                         float* __restrict__ out,
                         int nsamples)
{
    const int tid  = blockIdx.x * blockDim.x + threadIdx.x;
    const int lane = threadIdx.x & 31;
    const int s    = tid >> 5;
    if (s >= nsamples) return;              // wave-uniform branch: EXEC stays all-ones

    const int  d  = lane & 15;
    const bool hi = lane >= 16;

    const float* __restrict__ src =
        (hi ? B : A) + (((size_t)s * NDET + d) * (NDIM * NDIM));

    // ---- load this lane's 16x16 matrix into registers (64 x b128) ----
    float m[NDIM * NDIM];
    const float4* __restrict__ p4 = reinterpret_cast<const float4*>(src);
#pragma unroll
    for (int t = 0; t < 64; ++t) {
        float4 v = p4[t];
        m[4 * t + 0] = v.x;
        m[4 * t + 1] = v.y;
        m[4 * t + 2] = v.z;
        m[4 * t + 3] = v.w;
    }

    // ---- fully unrolled in-register LU (no pivoting) ----
    float    logdet = 0.0f;
    unsigned sbits  = 0u;
#pragma unroll
    for (int k = 0; k < NDIM; ++k) {
        const float p = m[k * NDIM + k];
        sbits  ^= (__float_as_uint(p) & 0x80000000u);
        logdet += __logf(fabsf(p));
        const float invp = __builtin_amdgcn_rcpf(p);
#pragma unroll
        for (int i = k + 1; i < NDIM; ++i) {
            const float mult = m[i * NDIM + k] * invp;
#pragma unroll
            for (int j = k + 1; j < NDIM; ++j) {
                m[i * NDIM + j] = fmaf(-mult, m[k * NDIM + j], m[i * NDIM + j]);
            }
        }
    }
    const float sgn = (sbits & 0x80000000u) ? -1.0f : 1.0f;

    // ---- combine a/b halves: lanes d and d+16 exchange their slogdets ----
    float x  = logdet + __shfl_xor(logdet, 16, 32);   // x_d = logdet_a + logdet_b
    float sg = sgn * __shfl_xor(sgn, 16, 32);         // sign_a * sign_b
    // (both 16-lane halves now hold identical 16-vectors indexed by d)

    // ---- xmax over the 16 determinants ----
    float xmax = x;
#pragma unroll
    for (int msk = 8; msk >= 1; msk >>= 1)
        xmax = fmaxf(xmax, __shfl_xor(xmax, msk, 32));

    const float e = sg * __expf(x - xmax);  // lane d (dup in d+16): sign_d * exp(x_d - xmax)

    // ---- weighted sum over dets on the f32 matrix pipe ----
    // D += A(16x4) x B(4x16), 4 chunks of K.  A-matrix f32 16x4 layout:
    //   lanes 0-15 : VGPR0 -> K=0, VGPR1 -> K=1
    //   lanes16-31 : VGPR0 -> K=2, VGPR1 -> K=3
    // All rows of A are set equal (e_{4kc+K}), B columns all equal w_{4kc+K},
    // so every element of D equals S = sum_d e_d * w_d.
    v8f acc = {0.f, 0.f, 0.f, 0.f, 0.f, 0.f, 0.f, 0.f};
#pragma unroll
    for (int kc = 0; kc < 4; ++kc) {
        const int k0 = 4 * kc + (hi ? 2 : 0);
        const int k1 = k0 + 1;
        v2f av, bv;
        av.x = __shfl(e, k0, 32);
        av.y = __shfl(e, k1, 32);
        bv.x = W[k0];
        bv.y = W[k1];
        acc = __builtin_amdgcn_wmma_f32_16x16x4_f32(
                  /*neg_a=*/false, av,
                  /*neg_b=*/false, bv,
                  /*c_mod=*/(short)0, acc,
                  /*reuse_a=*/false, /*reuse_b=*/false);
    }
    const float S = acc[0];   // identical in every lane / element

    const float log_psi  = __logf(fabsf(S)) + xmax;
    const float sign_psi = (S > 0.f) ? 1.f : ((S < 0.f) ? -1.f : 0.f);

    if (lane == 0) {
        out[s]            = log_psi;
        out[nsamples + s] = sign_psi;
    }
}

extern "C" void kernel_launch(void* const* d_in, const int* in_sizes, int n_in,
                              void* d_out, int out_size, void* d_ws, size_t ws_size,
                              hipStream_t stream) {
    (void)n_in; (void)d_ws; (void)ws_size;
    const float* a = (const float*)d_in[0];
    const float* b = (const float*)d_in[1];
    const float* w = (const float*)d_in[2];
    float* out = (float*)d_out;

    const int nsamples = out_size / 2;                 // 8192
    const int threads  = 128;                          // 4 waves/block, 1 sample/wave
    const int blocks   = (nsamples * 32 + threads - 1) / threads;

    logabssumdet_kernel<<<blocks, threads, 0, stream>>>(a, b, w, out, nsamples);
}